// TopKTopPSampler_53618371723375
// MI455X (gfx1250) — compile-verified
//
#include <hip/hip_runtime.h>

#define THREADS 512
#define CHUNK   4000   // floats per TDM chunk (16000 B); V=128000 -> 32 chunks

typedef __attribute__((ext_vector_type(4))) unsigned int u32x4;
typedef __attribute__((ext_vector_type(8))) int         i32x8;
typedef __attribute__((ext_vector_type(4))) int         i32x4;

static __device__ __forceinline__ float neg_inf() { return -__builtin_inff(); }

// Order-preserving map: float -> ascending unsigned key
static __device__ __forceinline__ unsigned fkey(float f) {
  unsigned u = __float_as_uint(f);
  return (u & 0x80000000u) ? ~u : (u | 0x80000000u);
}

// Issue one TDM descriptor: contiguous nElems f32 from global -> LDS.
// D# layout per CDNA5 ISA (group0: count/lds/global/type,
// group1: data_size=4B, tensor_dim0=tile_dim0=nElems, tensor_dim1=tile_dim1=1).
static __device__ __forceinline__ void tdm_load_chunk(const float* g, const float* ldsDst, int nElems) {
  unsigned long long ga = (unsigned long long)(size_t)g;
  unsigned ldsOff = (unsigned)(size_t)ldsDst;            // low 32b of flat shared addr = LDS offset
  u32x4 g0;
  g0[0] = 1u;                                            // count=1 (valid user descriptor)
  g0[1] = ldsOff;                                        // lds_addr
  g0[2] = (unsigned)(ga & 0xFFFFFFFFull);                // global_addr[31:0]
  g0[3] = (unsigned)((ga >> 32) & 0x1FFFFFFull) | (2u << 30); // global_addr[56:32] | type=2
  i32x8 g1;
  g1[0] = 0x00020000;                                    // wg_mask=0, data_size=2 (4 bytes)
  g1[1] = (nElems & 0xFFFF) << 16;                       // tensor_dim0[15:0]
  g1[2] = ((nElems >> 16) & 0xFFFF) | (1 << 16);         // tensor_dim0[31:16] | tensor_dim1=1
  g1[3] = (nElems & 0xFFFF) << 16;                       // tile_dim0 = nElems
  g1[4] = 1;                                             // tile_dim1=1, tile_dim2=0
  g1[5] = nElems;                                        // tensor_dim0_stride lo32
  g1[6] = 0;
  g1[7] = 0;
  i32x4 z4; z4[0] = 0; z4[1] = 0; z4[2] = 0; z4[3] = 0;  // groups 2/3 unused (<=2D tensor)
  i32x8 z8; z8[0] = 0; z8[1] = 0; z8[2] = 0; z8[3] = 0;
  z8[4] = 0; z8[5] = 0; z8[6] = 0; z8[7] = 0;
  // amdgpu-toolchain (clang-23) 6-arg form: (g0, g1, g2, g3, g4, cpol)
  __builtin_amdgcn_tensor_load_to_lds(g0, g1, z4, z4, z8, 0);
}

// Stream a row through LDS double-buffered via TDM; wave0 issues, all waves consume.
template <typename F>
static __device__ __forceinline__ void stream_row(const float* rowPtr, int V,
                                                  float (*stage)[CHUNK], bool issuer, F&& consume) {
  const int nChunks = (V + CHUNK - 1) / CHUNK;
  if (issuer) {
    int n0 = V < CHUNK ? V : CHUNK;
    tdm_load_chunk(rowPtr, stage[0], n0);
  }
  for (int c = 0; c < nChunks; ++c) {
    if (issuer) {
      if (c + 1 < nChunks) {
        int base1 = (c + 1) * CHUNK;
        int n1 = V - base1; if (n1 > CHUNK) n1 = CHUNK;
        tdm_load_chunk(rowPtr + base1, stage[(c + 1) & 1], n1);
        __builtin_amdgcn_s_wait_tensorcnt(1);   // chunk c landed
      } else {
        __builtin_amdgcn_s_wait_tensorcnt(0);   // last chunk landed
      }
    }
    __syncthreads();                            // publish LDS to all waves
    int base = c * CHUNK;
    int n = V - base; if (n > CHUNK) n = CHUNK;
    consume((const float*)stage[c & 1], base, n);
    __syncthreads();                            // done reading before buffer reuse
  }
}

__global__ __launch_bounds__(THREADS)
void topk_topp_sample_kernel(const float* __restrict__ logits,
                             const int*   __restrict__ kArr,
                             const float* __restrict__ pArr,
                             const float* __restrict__ qArr,
                             float* __restrict__ out,
                             int B, int V)
{
  const int row = blockIdx.x;
  const int t   = threadIdx.x;
  // Scalar branch condition: TDM ignores EXEC, so issuer-guard must be a real s_cbranch.
  const bool issuer = (__builtin_amdgcn_readfirstlane((int)(threadIdx.x >> 5)) == 0);

  const float* rowL = logits + (size_t)row * (size_t)V;
  const float* rowQ = qArr   + (size_t)row * (size_t)V;
  float*       rowO = out + B + (size_t)row * (size_t)V;

  __shared__ float hist[2048];
  __shared__ float aux[THREADS];
  __shared__ float stage[2][CHUNK];
  __shared__ int   sBin, sFlag;
  __shared__ float sCum, sMass, sMax;

  const int   kRow = kArr[row];
  const float pRow = pArr[row];
  const float kF   = (float)kRow;

  auto clearHist = [&](int nb) {
    for (int b = t; b < nb; b += THREADS) hist[b] = 0.f;
  };

  // Find unique bin d with cumAbove(d) < thresh <= cumAbove(d)+mass(d), scanning
  // bins in descending order. Returns false if total mass < thresh (keep all).
  auto scanCross = [&](int nbins, float thresh, int& bin, float& cum, float& mass, float& total) -> bool {
    const int per = nbins / THREADS;
    const int b0  = t * per;
    float s = 0.f;
    for (int j = 0; j < per; ++j) s += hist[b0 + j];
    aux[t] = s;
    __syncthreads();
    for (int st = 1; st < THREADS; st <<= 1) {   // inclusive suffix scan over groups
      float cur = aux[t];
      float v = (t + st < THREADS) ? aux[t + st] : 0.f;
      __syncthreads();
      aux[t] = cur + v;
      __syncthreads();
    }
    float tot = aux[0];
    if (t == 0) sFlag = 0;
    float sfx = (t + 1 < THREADS) ? aux[t + 1] : 0.f;  // mass strictly above this group
    __syncthreads();
    for (int j = per - 1; j >= 0; --j) {
      float h = hist[b0 + j];
      float cumAb = sfx;                               // mass strictly above bin b0+j
      sfx += h;
      if (h > 0.f && cumAb < thresh && thresh <= cumAb + h) {
        sBin = b0 + j; sCum = cumAb; sMass = h; sFlag = 1;
      }
    }
    __syncthreads();
    bin = sBin; cum = sCum; mass = sMass; total = tot;
    bool f = (sFlag != 0);
    __syncthreads();
    return f;
  };

  // ---------- Pass 1: row max + level-0 count histogram (key[31:21]) ----------
  clearHist(2048);
  float lmax = neg_inf();
  stream_row(rowL, V, stage, issuer, [&](const float* buf, int base, int n) {
    (void)base;
    for (int i = t; i < n; i += THREADS) {
      float x = buf[i];
      lmax = fmaxf(lmax, x);
      atomicAdd(&hist[fkey(x) >> 21], 1.f);
    }
  });
  aux[t] = lmax;
  __syncthreads();
  for (int s = THREADS / 2; s > 0; s >>= 1) {
    if (t < s) aux[t] = fmaxf(aux[t], aux[t + s]);
    __syncthreads();
  }
  if (t == 0) sMax = aux[0];
  __syncthreads();
  const float rowMax = sMax;
  __syncthreads();

  // ---------- Top-k radix select: 11 / 11 / 10 bits (counts exact in f32) ----------
  int bin; float cum, mass, total;
  scanCross(2048, kF, bin, cum, mass, total);
  const unsigned d0 = (unsigned)bin;
  float kRem = kF - cum;

  clearHist(2048);
  stream_row(rowL, V, stage, issuer, [&](const float* buf, int base, int n) {
    (void)base;
    for (int i = t; i < n; i += THREADS) {
      unsigned u = fkey(buf[i]);
      if ((u >> 21) == d0) atomicAdd(&hist[(u >> 10) & 0x7FFu], 1.f);
    }
  });
  scanCross(2048, kRem, bin, cum, mass, total);
  const unsigned d1 = (unsigned)bin;
  kRem -= cum;
  const unsigned pfx01 = (d0 << 11) | d1;

  clearHist(1024);
  stream_row(rowL, V, stage, issuer, [&](const float* buf, int base, int n) {
    (void)base;
    for (int i = t; i < n; i += THREADS) {
      unsigned u = fkey(buf[i]);
      if ((u >> 10) == pfx01) atomicAdd(&hist[u & 0x3FFu], 1.f);
    }
  });
  scanCross(1024, kRem, bin, cum, mass, total);
  const unsigned kthKey = (pfx01 << 10) | (unsigned)bin;  // key of k-th largest value

  // ---------- Top-p boundary: keep v iff (mass strictly above v) < p ----------
  unsigned cutKey = kthKey;
  float M;  // final kept probability mass, relative to exp(rowMax)
  clearHist(2048);
  stream_row(rowL, V, stage, issuer, [&](const float* buf, int base, int n) {
    (void)base;
    for (int i = t; i < n; i += THREADS) {
      float x = buf[i];
      unsigned u = fkey(x);
      if (u >= kthKey) atomicAdd(&hist[u >> 21], __expf(x - rowMax));
    }
  });
  bool found = scanCross(2048, pRow, bin, cum, mass, total);
  M = total;                                   // no crossing -> keep whole top-k set
  if (found) {
    const unsigned e0 = (unsigned)bin;
    const float base1 = cum;
    clearHist(2048);
    stream_row(rowL, V, stage, issuer, [&](const float* buf, int base, int n) {
      (void)base;
      for (int i = t; i < n; i += THREADS) {
        float x = buf[i];
        unsigned u = fkey(x);
        if (u >= kthKey && (u >> 21) == e0) atomicAdd(&hist[(u >> 10) & 0x7FFu], __expf(x - rowMax));
      }
    });
    found = scanCross(2048, pRow - base1, bin, cum, mass, total);
    if (!found) {
      unsigned c0 = e0 << 21;
      cutKey = (c0 > kthKey) ? c0 : kthKey;
      M = base1 + total;
    } else {
      const unsigned e1 = (unsigned)bin;
      const float base2 = base1 + cum;
      const unsigned ppfx = (e0 << 11) | e1;
      clearHist(1024);
      stream_row(rowL, V, stage, issuer, [&](const float* buf, int base, int n) {
        (void)base;
        for (int i = t; i < n; i += THREADS) {
          float x = buf[i];
          unsigned u = fkey(x);
          if (u >= kthKey && (u >> 10) == ppfx) atomicAdd(&hist[u & 0x3FFu], __expf(x - rowMax));
        }
      });
      found = scanCross(1024, pRow - base2, bin, cum, mass, total);
      if (!found) {
        unsigned c0 = ppfx << 10;
        cutKey = (c0 > kthKey) ? c0 : kthKey;
        M = base2 + total;
      } else {
        unsigned bKey = (ppfx << 10) | (unsigned)bin;   // boundary element itself is kept
        cutKey = (bKey > kthKey) ? bKey : kthKey;
        M = base2 + cum + mass;
      }
    }
  }

  // ---------- Final pass: log-softmax write + exponential-trick argmax ----------
  const float    lse  = rowMax + __logf(M);
  const unsigned cutK = cutKey;
  float bestV = neg_inf();
  int   bestI = 0;
  stream_row(rowL, V, stage, issuer, [&](const float* buf, int base, int n) {
    __builtin_prefetch(rowQ + base, 0, 1);     // global_prefetch for the q stream
    for (int i = t; i < n; i += THREADS) {
      int gi = base + i;
      float x = buf[i];
      bool kept = fkey(x) >= cutK;
      rowO[gi] = kept ? (x - lse) : neg_inf();
      // argmax(probs/q) == argmax over kept of (x - log q)
      float sc = kept ? (x - __logf(rowQ[gi])) : neg_inf();
      if (sc > bestV) { bestV = sc; bestI = gi; }
    }
  });
  // Block argmax reduce, first index wins ties
  int* idxBuf = (int*)hist;
  aux[t] = bestV; idxBuf[t] = bestI;
  __syncthreads();
  for (int s = THREADS / 2; s > 0; s >>= 1) {
    if (t < s) {
      float v2 = aux[t + s]; int i2 = idxBuf[t + s];
      if (v2 > aux[t] || (v2 == aux[t] && i2 < idxBuf[t])) { aux[t] = v2; idxBuf[t] = i2; }
    }
    __syncthreads();
  }
  if (t == 0) out[row] = (float)idxBuf[0];
}

extern "C" void kernel_launch(void* const* d_in, const int* in_sizes, int n_in,
                              void* d_out, int out_size, void* d_ws, size_t ws_size,
                              hipStream_t stream) {
  (void)n_in; (void)out_size; (void)d_ws; (void)ws_size;
  const float* logits = (const float*)d_in[0];
  const int*   k      = (const int*)d_in[1];
  const float* p      = (const float*)d_in[2];
  const float* q      = (const float*)d_in[3];
  const int B = in_sizes[1];                 // 128 rows (k is per-row)
  const int V = in_sizes[0] / B;             // 128000
  topk_topp_sample_kernel<<<B, THREADS, 0, stream>>>(logits, k, p, q, (float*)d_out, B, V);
}